// MultiDiffusion_81844896792601
// MI455X (gfx1250) — compile-verified
//
#include <hip/hip_runtime.h>
#include <hip/hip_bf16.h>

typedef __attribute__((ext_vector_type(2))) float v2f;
typedef __attribute__((ext_vector_type(8))) float v8f;

#define OBS 4
#define HID 64
#define FEAT (OBS * HID)   // 256 floats per node row
#define K_HOPS 5

// ---------------------------------------------------------------------------
// X[n*FEAT + o*HID + h] = nodes[n*OBS+o] * W_lift[h] + b_lift[h]
// ---------------------------------------------------------------------------
__global__ void lift_kernel(const float* __restrict__ nodes,
                            const float* __restrict__ Wl,
                            const float* __restrict__ bl,
                            float* __restrict__ X, int total /* N*OBS*HID */) {
    int t = blockIdx.x * blockDim.x + threadIdx.x;
    if (t >= total) return;
    int h  = t & (HID - 1);
    int no = t >> 6;                 // index into (N, OBS)
    X[t] = nodes[no] * Wl[h] + bl[h];
}

// ---------------------------------------------------------------------------
// zero agg buffer (float4 stores)
// ---------------------------------------------------------------------------
__global__ void zero_kernel(float4* __restrict__ p, int n4) {
    int t = blockIdx.x * blockDim.x + threadIdx.x;
    if (t < n4) p[t] = make_float4(0.f, 0.f, 0.f, 0.f);
}

// ---------------------------------------------------------------------------
// One block (256 threads) per edge. Edge scalars come from blockIdx -> SGPR
// loads. Gather X[src] row (coalesced), scale, no-return float atomic into
// agg[dst] row. agg/X are L2-resident (51.2MB each vs 192MB L2).
// ---------------------------------------------------------------------------
__global__ void scatter_kernel(const int* __restrict__ esrc,
                               const int* __restrict__ edst,
                               const float* __restrict__ ew,
                               const float* __restrict__ X,
                               float* __restrict__ agg) {
    int e = blockIdx.x;
    int s = esrc[e];
    int d = edst[e];
    float w = ew[e];
    int f = threadIdx.x;             // 0..255
    float val = w * X[(size_t)s * FEAT + f];
    // relaxed, agent scope, result discarded -> no-return f32 atomic
    __hip_atomic_fetch_add(&agg[(size_t)d * FEAT + f], val,
                           __ATOMIC_RELAXED, __HIP_MEMORY_SCOPE_AGENT);
}

// ---------------------------------------------------------------------------
// out = relu(agg @ W + b), agg: (rows, 64), W: (64, 64) row-major (in, out).
// One wave per 16-row tile; 4 v8f accumulators cover all 64 output columns.
// K swept in 16 steps of V_WMMA_F32_16X16X4_F32.
//
// f32 16x16x4 fragment layout (ISA 7.12.2):
//   A (16x4): lane m = l&15; VGPR0 = A[m][khalf+0], VGPR1 = A[m][khalf+1],
//             khalf = (l<16) ? 0 : 2
//   B (4x16): lane n = l&15; VGPR0 = B[khalf+0][n], VGPR1 = B[khalf+1][n]
//   C/D (16x16): VGPR v: lanes 0-15 -> (M=v, N=l), lanes 16-31 -> (M=v+8)
//
// W is staged in LDS packed as K-pairs: sWp[p*64 + n] = {W[2p][n], W[2p+1][n]}
// so each B fragment is a single aligned ds_load_b64 landing in a contiguous
// even-aligned VGPR pair (no marshaling movs before the WMMA).
// ---------------------------------------------------------------------------
__global__ void gemm_relu_kernel(const float* __restrict__ agg,
                                 const float* __restrict__ W,
                                 const float* __restrict__ bias,
                                 float* __restrict__ out, int rows) {
    __shared__ v2f   sWp[(HID / 2) * HID];   // 32 pair-rows x 64 cols, 16 KB
    __shared__ float sB[HID];

    // pack: i = p*64 + n  ->  {W[2p][n], W[2p+1][n]}
    for (int i = threadIdx.x; i < (HID / 2) * HID; i += blockDim.x) {
        int p = i >> 6;
        int n = i & 63;
        v2f w2;
        w2.x = W[(2 * p) * HID + n];
        w2.y = W[(2 * p + 1) * HID + n];
        sWp[i] = w2;
    }
    if (threadIdx.x < HID) sB[threadIdx.x] = bias[threadIdx.x];
    __syncthreads();

    const int wave = threadIdx.x >> 5;          // 0..7
    const int lane = threadIdx.x & 31;
    const int tile = blockIdx.x * 8 + wave;     // 16-row tile index
    const int row0 = tile * 16;
    if (row0 >= rows) return;                   // wave-uniform: EXEC stays all-1s

    const int mlane = lane & 15;
    const int khalf = (lane < 16) ? 0 : 2;

    v8f acc0 = (v8f){0,0,0,0,0,0,0,0};
    v8f acc1 = acc0, acc2 = acc0, acc3 = acc0;

    const float* arow = agg + (size_t)(row0 + mlane) * HID;
    // per-lane base into packed W rows for this lane's K-half
    const v2f* wbase = &sWp[(khalf >> 1) * HID + mlane];

    #pragma unroll
    for (int k = 0; k < HID; k += 4) {
        // A fragment: two consecutive f32 (8B aligned: k+khalf is even)
        v2f a = *(const v2f*)(arow + k + khalf);
        const v2f* wp = wbase + (k >> 1) * HID;  // pair-row (k+khalf)/2

        v2f b0 = wp[0];
        acc0 = __builtin_amdgcn_wmma_f32_16x16x4_f32(false, a, false, b0,
                                                     (short)0, acc0, false, false);
        v2f b1 = wp[16];
        acc1 = __builtin_amdgcn_wmma_f32_16x16x4_f32(false, a, false, b1,
                                                     (short)0, acc1, false, false);
        v2f b2 = wp[32];
        acc2 = __builtin_amdgcn_wmma_f32_16x16x4_f32(false, a, false, b2,
                                                     (short)0, acc2, false, false);
        v2f b3 = wp[48];
        acc3 = __builtin_amdgcn_wmma_f32_16x16x4_f32(false, a, false, b3,
                                                     (short)0, acc3, false, false);
    }

    // Write back with bias + ReLU
    const int mofs = (lane < 16) ? 0 : 8;
    v8f accs[4] = {acc0, acc1, acc2, acc3};
    #pragma unroll
    for (int t = 0; t < 4; t++) {
        const int ncol = t * 16 + mlane;
        const float bv = sB[ncol];
        #pragma unroll
        for (int v = 0; v < 8; v++) {
            const int m = row0 + mofs + v;
            float val = accs[t][v] + bv;
            out[(size_t)m * HID + ncol] = val > 0.f ? val : 0.f;
        }
    }
}

// ---------------------------------------------------------------------------
// Driver: X lives in d_out (same size as the result); agg in d_ws.
//   lift -> [zero agg; scatter; gemm_relu] x 5
// ---------------------------------------------------------------------------
extern "C" void kernel_launch(void* const* d_in, const int* in_sizes, int n_in,
                              void* d_out, int out_size, void* d_ws, size_t ws_size,
                              hipStream_t stream) {
    const float* nodes = (const float*)d_in[0];   // (N, OBS)
    const int*   esrc  = (const int*)  d_in[1];   // (E,)
    const int*   edst  = (const int*)  d_in[2];   // (E,)
    const float* ew    = (const float*)d_in[3];   // (E,)
    const float* Wl    = (const float*)d_in[4];   // (1, HID)
    const float* bl    = (const float*)d_in[5];   // (HID,)
    const float* Wh    = (const float*)d_in[6];   // (HID, HID)
    const float* bh    = (const float*)d_in[7];   // (HID,)

    const int E     = in_sizes[1];
    const int N     = in_sizes[0] / OBS;
    const int rows  = N * OBS;            // 200000
    const int total = rows * HID;         // 12.8M floats

    float* X   = (float*)d_out;           // ping buffer == final output
    float* agg = (float*)d_ws;            // 51.2 MB scratch

    // 1) lift
    lift_kernel<<<dim3((total + 255) / 256), dim3(256), 0, stream>>>(
        nodes, Wl, bl, X, total);

    const int n4        = total / 4;
    const int zeroBlks  = (n4 + 255) / 256;
    const int tiles     = rows / 16;                   // 12500
    const int gemmBlks  = (tiles + 7) / 8;             // 1563

    for (int hop = 0; hop < K_HOPS; ++hop) {
        zero_kernel<<<dim3(zeroBlks), dim3(256), 0, stream>>>((float4*)agg, n4);
        scatter_kernel<<<dim3(E), dim3(256), 0, stream>>>(esrc, edst, ew, X, agg);
        gemm_relu_kernel<<<dim3(gemmBlks), dim3(256), 0, stream>>>(
            agg, Wh, bh, X, rows);
    }
}